// KAConvTritonVectorized_78056735637899
// MI455X (gfx1250) — compile-verified
//
#include <hip/hip_runtime.h>

// KAConv (per-weight Pade rational over conv windows) for MI455X / gfx1250.
// Core compute mapped onto V_WMMA_F32_16X16X4_F32:
//   per (q, c):  num[16pix x 16f] = [1 w w2 w3 | w4 w5 0 0] x coeff rows (2 WMMAs)
//                den[16pix x 16f] = [w w2 w3 w4]            x coeff rows (1 WMMA)
//   acc += num * rcp(1 + |den|)   (elementwise VALU on the 8 accum VGPRs)
// Block = 8 waves; waves split channels, LDS tree-reduces the 16x16 tile.

typedef __attribute__((ext_vector_type(2))) float v2f;
typedef __attribute__((ext_vector_type(4))) float v4f;
typedef __attribute__((ext_vector_type(8))) float v8f;

#define NB   8
#define NC   32
#define NF   32
#define NH   32
#define NW   32
#define NQ   9
#define NHW  (NH * NW)          // 1024
#define NPIX (NB * NHW)         // 8192

__global__ __launch_bounds__(256)
void kaconv_pade_wmma_kernel(const float* __restrict__ x,
                             const float* __restrict__ nums,
                             const float* __restrict__ denoms,
                             float* __restrict__ out)
{
    __shared__ float red[8 * 256];          // 8 waves x (16 pix x 16 f) partials

    const int  lane  = threadIdx.x & 31;
    const int  wv    = threadIdx.x >> 5;    // wave in block -> channel group
    const int  tile  = blockIdx.x >> 1;     // 16-pixel tile (0..511)
    const int  ftile = blockIdx.x & 1;      // filter half (0..1)
    const int  fbase = ftile * 16;
    const int  m     = lane & 15;           // matrix row / column for this lane
    const bool hif   = lane >= 16;          // high half-wave: K=2,3 slots

    // Pixel tile is 16 consecutive flat pixels -> same b, same y for all lanes.
    const int pix0 = tile * 16;
    const int b    = pix0 >> 10;            // / NHW
    const int yx0  = pix0 & (NHW - 1);
    const int y    = (yx0 + m) >> 5;
    const int xcol = (yx0 + m) & 31;

    const int f     = fbase + m;            // B-matrix column = filter
    const int cbase = wv * (NC / 8);        // 4 channels per wave

    v8f acc = {0.f, 0.f, 0.f, 0.f, 0.f, 0.f, 0.f, 0.f};
    const int off2 = hif ? 2 : 0;           // K2/K3 coefficient slot offset

    for (int q = 0; q < NQ; ++q) {
        const int  dy = q / 3 - 1;
        const int  dx = q % 3 - 1;
        const int  yy = y + dy;
        const int  xx = xcol + dx;
        const bool inb = (yy >= 0) & (yy < NH) & (xx >= 0) & (xx < NW);

        const float* xp  = x + ((b * NC + cbase) * NHW) + yy * NW + xx;
        const int    idx = (f * NQ + q) * NC + cbase;   // coeff row index
        const float* np  = nums   + (size_t)idx * 6;
        const float* dp  = denoms + (size_t)idx * 4;

        if (q + 1 < NQ) {                   // warm L2/L1 for next q's coeffs
            __builtin_prefetch(nums   + (size_t)((f * NQ + q + 1) * NC + cbase) * 6, 0, 1);
            __builtin_prefetch(denoms + (size_t)((f * NQ + q + 1) * NC + cbase) * 4, 0, 1);
        }

        #pragma unroll
        for (int c = 0; c < NC / 8; ++c) {
            // ---- A operands: per-lane powers of the window value -------------
            const float w  = inb ? xp[c * NHW] : 0.0f;
            const float w2 = w * w;
            const float w3 = w2 * w;
            const float w4 = w2 * w2;
            const float w5 = w4 * w;

            const v2f a_lo  = hif ? (v2f){w2, w3}  : (v2f){1.0f, w};   // p0..p3
            const v2f a_hi  = hif ? (v2f){0.f, 0.f} : (v2f){w4, w5};   // p4,p5
            const v2f a_den = hif ? (v2f){w3, w4}  : (v2f){w, w2};     // p1..p4

            // ---- B operands: coefficient rows (aligned b64 loads) ------------
            const float* n = np + c * 6;
            const float* d = dp + c * 4;
            const v2f bn_lo = *(const v2f*)(n + off2);  // p{0,1} / p{2,3}
            const v2f bn_hi = *(const v2f*)(n + 4);     // p{4,5} (K2/K3 x 0)
            const v2f bd    = *(const v2f*)(d + off2);  // p{1,2} / p{3,4}

            // ---- 3x V_WMMA_F32_16X16X4_F32 ----------------------------------
            const v8f z = {0.f, 0.f, 0.f, 0.f, 0.f, 0.f, 0.f, 0.f};
            v8f num = __builtin_amdgcn_wmma_f32_16x16x4_f32(
                          false, a_lo,  false, bn_lo, (short)0, z,   false, false);
            num     = __builtin_amdgcn_wmma_f32_16x16x4_f32(
                          false, a_hi,  false, bn_hi, (short)0, num, false, false);
            v8f den = __builtin_amdgcn_wmma_f32_16x16x4_f32(
                          false, a_den, false, bd,    (short)0, z,   false, false);

            // ---- acc += num / (1 + |den|) -----------------------------------
            #pragma unroll
            for (int i = 0; i < 8; ++i) {
                const float r = __builtin_amdgcn_rcpf(1.0f + __builtin_fabsf(den[i]));
                acc[i] = __builtin_fmaf(num[i], r, acc[i]);
            }
        }
    }

    // ---- cross-wave (channel-group) reduction through LDS -------------------
    const int mrow = hif ? 8 : 0;           // C layout: VGPR i -> M = i (+8 hi)
    #pragma unroll
    for (int i = 0; i < 8; ++i)
        red[wv * 256 + (i + mrow) * 16 + m] = acc[i];
    __syncthreads();

    // 256 threads == 256 tile elements: thread t owns (M = t>>4, N = t&15)
    const int t = threadIdx.x;
    float s = 0.0f;
    #pragma unroll
    for (int g = 0; g < 8; ++g)
        s += red[g * 256 + t];

    const int M = t >> 4;
    const int N = t & 15;
    out[(size_t)b * NF * NHW + (size_t)(fbase + N) * NHW + yx0 + M] = s;
}

extern "C" void kernel_launch(void* const* d_in, const int* in_sizes, int n_in,
                              void* d_out, int out_size, void* d_ws, size_t ws_size,
                              hipStream_t stream) {
    const float* x      = (const float*)d_in[0];   // [8,32,32,32]
    const float* nums   = (const float*)d_in[1];   // [F*Q*C, 6]
    const float* denoms = (const float*)d_in[2];   // [F*Q*C, 4]
    float*       out    = (float*)d_out;           // [8,32,32,32]

    // (NPIX/16) pixel tiles x 2 filter tiles = 1024 blocks, 256 threads (8 waves)
    dim3 grid((NPIX / 16) * 2);
    dim3 block(256);
    kaconv_pade_wmma_kernel<<<grid, block, 0, stream>>>(x, nums, denoms, out);
}